// OT_Head_20229295965011
// MI455X (gfx1250) — compile-verified
//
#include <hip/hip_runtime.h>
#include <cstdint>

#define EPS_F 1e-8f

typedef float v2f_t __attribute__((ext_vector_type(2)));
typedef float v8f_t __attribute__((ext_vector_type(8)));

// ---------------------------------------------------------------- reductions
__device__ __forceinline__ float wave_sum(float v) {
#pragma unroll
  for (int off = 16; off > 0; off >>= 1) v += __shfl_xor(v, off, 32);
  return v;
}

__device__ __forceinline__ float block_sum256(float v, float* sbuf) {
  int t = threadIdx.x;
  sbuf[t] = v;
  __syncthreads();
  for (int s = 128; s > 0; s >>= 1) {
    if (t < s) sbuf[t] += sbuf[t + s];
    __syncthreads();
  }
  float r = sbuf[0];
  __syncthreads();
  return r;
}

// ---------------------------------------------------------------- 1) GEMM
// scores[b,m,n] = sum_d se[b,d,m]*te[b,d,n] * scale, via V_WMMA_F32_16X16X4_F32.
// Block = 128 threads (4 waves); block tile 64x64.
// k-chunks of 8 are staged in LDS with coalesced b128 loads (1 float4 of A and
// 1 of B per thread per chunk), killing the 4x redundant A fetches across the
// block's 4 waves; WMMA fragments then come from ds_load. Next chunk is
// prefetched with global_prefetch_b8.
#define GEMM_KC 8
__global__ __launch_bounds__(128) void gemm_scores_k(
    const float* __restrict__ se, const float* __restrict__ te,
    float* __restrict__ scores, int D, int M, int N, float scale) {
  __shared__ float As[GEMM_KC][64];  // As[d][m]
  __shared__ float Bs[GEMM_KC][64];  // Bs[d][n]
  const int b     = blockIdx.z;
  const int m0    = blockIdx.x * 64;
  const int n0blk = blockIdx.y * 64;
  const int tid   = threadIdx.x;
  const int wave  = tid >> 5;
  const int lane  = tid & 31;
  const int nw    = wave * 16;        // this wave's N offset within the tile
  const int ml    = lane & 15;
  const int kh    = (lane >> 4) << 1; // 0 or 2 (A/B fragment k-halves)
  const int ld_d  = tid >> 4;         // 0..7: staged k-row
  const int ld_m  = (tid & 15) * 4;   // 0..60: float4 column
  const float* A  = se + (size_t)b * D * M;  // A[d*M + m]
  const float* Bp = te + (size_t)b * D * N;  // B[d*N + n]
  v8f_t acc0 = {}, acc1 = {}, acc2 = {}, acc3 = {};
  for (int k0 = 0; k0 < D; k0 += GEMM_KC) {
    __syncthreads();  // protect LDS from previous stage's readers
    const float4 av = *(const float4*)(A + (size_t)(k0 + ld_d) * M + (m0 + ld_m));
    const float4 bv = *(const float4*)(Bp + (size_t)(k0 + ld_d) * N + (n0blk + ld_m));
    *(float4*)&As[ld_d][ld_m] = av;
    *(float4*)&Bs[ld_d][ld_m] = bv;
    if (k0 + GEMM_KC < D) {  // prefetch next chunk (global_prefetch_b8)
      __builtin_prefetch(A + (size_t)(k0 + GEMM_KC + ld_d) * M + (m0 + ld_m), 0, 1);
      __builtin_prefetch(Bp + (size_t)(k0 + GEMM_KC + ld_d) * N + (n0blk + ld_m), 0, 1);
    }
    __syncthreads();
#pragma unroll
    for (int kk = 0; kk < GEMM_KC; kk += 4) {
      v2f_t bf;
      bf.x = Bs[kk + kh][nw + ml];
      bf.y = Bs[kk + kh + 1][nw + ml];
      v2f_t a0, a1, a2, a3;
      a0.x = As[kk + kh][ml];      a0.y = As[kk + kh + 1][ml];
      a1.x = As[kk + kh][16 + ml]; a1.y = As[kk + kh + 1][16 + ml];
      a2.x = As[kk + kh][32 + ml]; a2.y = As[kk + kh + 1][32 + ml];
      a3.x = As[kk + kh][48 + ml]; a3.y = As[kk + kh + 1][48 + ml];
      acc0 = __builtin_amdgcn_wmma_f32_16x16x4_f32(false, a0, false, bf, (short)0, acc0, false, false);
      acc1 = __builtin_amdgcn_wmma_f32_16x16x4_f32(false, a1, false, bf, (short)0, acc1, false, false);
      acc2 = __builtin_amdgcn_wmma_f32_16x16x4_f32(false, a2, false, bf, (short)0, acc2, false, false);
      acc3 = __builtin_amdgcn_wmma_f32_16x16x4_f32(false, a3, false, bf, (short)0, acc3, false, false);
    }
  }
  const int rb = (lane >> 4) * 8;  // C/D layout: VGPR r -> row r (lanes 0-15), row 8+r (16-31)
  const int n  = n0blk + nw + ml;
  const size_t rowbase = (size_t)b * M;
#pragma unroll
  for (int r = 0; r < 8; ++r) {
    scores[(rowbase + m0 +      rb + r) * (size_t)N + n] = acc0[r] * scale;
    scores[(rowbase + m0 + 16 + rb + r) * (size_t)N + n] = acc1[r] * scale;
    scores[(rowbase + m0 + 32 + rb + r) * (size_t)N + n] = acc2[r] * scale;
    scores[(rowbase + m0 + 48 + rb + r) * (size_t)N + n] = acc3[r] * scale;
  }
}

// ---------------------------------------------------------------- 2) row logsumexp
__global__ __launch_bounds__(256) void row_lse_k(const float* __restrict__ S,
                                                 float* __restrict__ lse, int N) {
  const int wid  = (blockIdx.x * blockDim.x + threadIdx.x) >> 5;  // row over B*M
  const int lane = threadIdx.x & 31;
  const float* row = S + (size_t)wid * N;
  float mx = -3.4e38f, sm = 0.f;
  for (int j = lane; j < N; j += 32) {
    float x = row[j];
    if (x > mx) { sm = sm * __expf(mx - x) + 1.f; mx = x; }
    else        { sm += __expf(x - mx); }
  }
#pragma unroll
  for (int off = 16; off > 0; off >>= 1) {
    float m2 = __shfl_xor(mx, off, 32);
    float s2 = __shfl_xor(sm, off, 32);
    float mn = fmaxf(mx, m2);
    sm = sm * __expf(mx - mn) + s2 * __expf(m2 - mn);
    mx = mn;
  }
  if (lane == 0) lse[wid] = mx + __logf(sm);
}

// ---------------------------------------------------------------- 3) K0 = exp((s-lse)/rho), row partial sums
__global__ __launch_bounds__(256) void exp_rows_k(float* __restrict__ K,
                                                  const float* __restrict__ lse,
                                                  const float* __restrict__ rho,
                                                  float* __restrict__ rowpart, int N) {
  const int row = blockIdx.x;  // over B*M
  const float l = lse[row];
  const float ir = 1.f / fmaxf(rho[0], EPS_F);
  float* r = K + (size_t)row * N;
  float p = 0.f;
  for (int j = threadIdx.x; j < N; j += 256) {
    float kv = __expf((r[j] - l) * ir);
    r[j] = kv;
    p += kv;
  }
  __shared__ float sbuf[256];
  float s = block_sum256(p, sbuf);
  if (threadIdx.x == 0) rowpart[row] = s;
}

__global__ __launch_bounds__(256) void g_init_k(const float* __restrict__ rowpart,
                                                float* __restrict__ g, int M) {
  const int b = blockIdx.x;
  float p = 0.f;
  for (int i = threadIdx.x; i < M; i += 256) p += rowpart[b * M + i];
  __shared__ float sbuf[256];
  float s = block_sum256(p, sbuf);
  if (threadIdx.x == 0) g[b] = 1.f / fmaxf(s, EPS_F);  // mass m = 1
}

__global__ void init_uv_k(float* __restrict__ u, float* __restrict__ v, int BM, int BN) {
  const int idx = blockIdx.x * blockDim.x + threadIdx.x;
  if (idx < BM) u[idx] = 1.f;
  if (idx < BN) v[idx] = 1.f;
}

// ---------------------------------------------------------------- Sinkhorn matvecs
__global__ __launch_bounds__(256) void rowmv_k(const float* __restrict__ K,
                                               const float* __restrict__ v,
                                               float* __restrict__ y, int M, int N) {
  const int wid  = (blockIdx.x * blockDim.x + threadIdx.x) >> 5;  // row over B*M
  const int lane = threadIdx.x & 31;
  const int b = wid / M;
  const float* row = K + (size_t)wid * N;
  const float* vb  = v + (size_t)b * N;
  float acc = 0.f;
  for (int j = lane; j < N; j += 32) acc = fmaf(row[j], vb[j], acc);
  acc = wave_sum(acc);
  if (lane == 0) y[wid] = acc;
}

__global__ void u_update_k(float* __restrict__ u, const float* __restrict__ y,
                           const float* __restrict__ g, float aw, int M, int BM) {
  const int idx = blockIdx.x * blockDim.x + threadIdx.x;
  if (idx >= BM) return;
  const int b = idx / M;
  float rs = g[b] * u[idx] * y[idx];                 // rowsum of current K
  float r  = fminf(aw / fmaxf(rs, EPS_F), 1.f);      // min(a/clip(rowsum),1)
  u[idx] *= r;
}

__global__ __launch_bounds__(256) void colmv_part_k(const float* __restrict__ K,
                                                    const float* __restrict__ u,
                                                    float* __restrict__ zpart,
                                                    int M, int N, int CCH) {
  const int b  = blockIdx.z;
  const int ch = blockIdx.y;
  const int j  = blockIdx.x * 256 + threadIdx.x;
  const int i0 = ch * (M / CCH), i1 = i0 + (M / CCH);
  const float* Kb = K + (size_t)b * M * N;
  const float* ub = u + (size_t)b * M;
  float acc = 0.f;
  for (int i = i0; i < i1; ++i) acc = fmaf(Kb[(size_t)i * N + j], ub[i], acc);
  zpart[((size_t)(b * CCH + ch)) * N + j] = acc;
}

__global__ void z_reduce_k(const float* __restrict__ zpart, float* __restrict__ z,
                           int N, int CCH, int BN) {
  const int idx = blockIdx.x * blockDim.x + threadIdx.x;
  if (idx >= BN) return;
  const int b = idx / N, j = idx % N;
  float s = 0.f;
  for (int c = 0; c < CCH; ++c) s += zpart[((size_t)(b * CCH + c)) * N + j];
  z[idx] = s;
}

__global__ __launch_bounds__(256) void v_update_k(float* __restrict__ v,
                                                  const float* __restrict__ z,
                                                  const float* __restrict__ g,
                                                  float* __restrict__ Spart,
                                                  float bw, int N) {
  const int b = blockIdx.y;
  const int j = blockIdx.x * 256 + threadIdx.x;
  const size_t idx = (size_t)b * N + j;
  const float gb = g[b];
  float zv = z[idx];
  float cs = gb * v[idx] * zv;                       // colsum of K1
  float c  = fminf(bw / fmaxf(cs, EPS_F), 1.f);
  float vn = v[idx] * c;
  v[idx] = vn;
  __shared__ float sbuf[256];
  float sp = block_sum256(gb * vn * zv, sbuf);       // contribution to sum(K2)
  if (threadIdx.x == 0) Spart[b * gridDim.x + blockIdx.x] = sp;
}

__global__ void g_update_k(float* __restrict__ g, const float* __restrict__ Spart, int nblk) {
  const int b = threadIdx.x;
  float S = 0.f;
  for (int k = 0; k < nblk; ++k) S += Spart[b * nblk + k];
  g[b] *= 1.f / fmaxf(S, EPS_F);                     // K *= m/clip(sum)
}

// ---------------------------------------------------------------- finalize matches + weighted_ref
__global__ __launch_bounds__(256) void finalize_rows_k(
    float* __restrict__ K, const float* __restrict__ u, const float* __restrict__ v,
    const float* __restrict__ g, const float* __restrict__ tgt,
    float* __restrict__ rowsum, float* __restrict__ wref, int M, int N) {
  const int wid  = (blockIdx.x * blockDim.x + threadIdx.x) >> 5;
  const int lane = threadIdx.x & 31;
  const int b = wid / M;
  const float gu = g[b] * u[wid];
  float* row = K + (size_t)wid * N;
  const float* vb = v + (size_t)b * N;
  const float* t0 = tgt + (size_t)b * 3 * N;
  float s = 0.f, w0 = 0.f, w1 = 0.f, w2 = 0.f;
  for (int j = lane; j < N; j += 32) {
    float mv = gu * row[j] * vb[j];
    row[j] = mv;                                     // final matches, in place
    s += mv;
    w0 = fmaf(mv, t0[j], w0);
    w1 = fmaf(mv, t0[N + j], w1);
    w2 = fmaf(mv, t0[2 * N + j], w2);
  }
  s  = wave_sum(s);
  w0 = wave_sum(w0);
  w1 = wave_sum(w1);
  w2 = wave_sum(w2);
  if (lane == 0) {
    rowsum[wid] = s;
    float inv = 1.f / (s + 1e-6f);
    wref[(size_t)wid * 3 + 0] = w0 * inv;
    wref[(size_t)wid * 3 + 1] = w1 * inv;
    wref[(size_t)wid * 3 + 2] = w2 * inv;
  }
}

// ---------------------------------------------------------------- rigid transform (per batch)
__global__ __launch_bounds__(256) void rigid_k(const float* __restrict__ src,
                                               const float* __restrict__ wref,
                                               const float* __restrict__ w,
                                               float* __restrict__ outR,
                                               float* __restrict__ outT, int M) {
  const int b = blockIdx.x;
  const int t = threadIdx.x;
  __shared__ float sbuf[256];
  const float* s3 = src  + (size_t)b * 3 * M;   // src[b,k,i]
  const float* wr = wref + (size_t)b * M * 3;   // (M,3)
  const float* wb = w    + (size_t)b * M;
  // pass 1: weights & weighted centroids
  float psw = 0.f, pa0 = 0.f, pa1 = 0.f, pa2 = 0.f, pb0 = 0.f, pb1 = 0.f, pb2 = 0.f;
  for (int i = t; i < M; i += 256) {
    float wi = wb[i];
    psw += wi;
    pa0 = fmaf(s3[i], wi, pa0);
    pa1 = fmaf(s3[M + i], wi, pa1);
    pa2 = fmaf(s3[2 * M + i], wi, pa2);
    pb0 = fmaf(wr[(size_t)i * 3 + 0], wi, pb0);
    pb1 = fmaf(wr[(size_t)i * 3 + 1], wi, pb1);
    pb2 = fmaf(wr[(size_t)i * 3 + 2], wi, pb2);
  }
  float sw  = block_sum256(psw, sbuf);
  float inv = 1.f / (sw + 1e-6f);
  float ca[3], cb[3];
  ca[0] = block_sum256(pa0, sbuf) * inv;
  ca[1] = block_sum256(pa1, sbuf) * inv;
  ca[2] = block_sum256(pa2, sbuf) * inv;
  cb[0] = block_sum256(pb0, sbuf) * inv;
  cb[1] = block_sum256(pb1, sbuf) * inv;
  cb[2] = block_sum256(pb2, sbuf) * inv;
  // pass 2: cov = ac^T (bc * wn)
  float pc[9] = {0.f, 0.f, 0.f, 0.f, 0.f, 0.f, 0.f, 0.f, 0.f};
  for (int i = t; i < M; i += 256) {
    float wn = wb[i] * inv;
    float ax = s3[i] - ca[0], ay = s3[M + i] - ca[1], az = s3[2 * M + i] - ca[2];
    float bx = (wr[(size_t)i * 3 + 0] - cb[0]) * wn;
    float by = (wr[(size_t)i * 3 + 1] - cb[1]) * wn;
    float bz = (wr[(size_t)i * 3 + 2] - cb[2]) * wn;
    pc[0] = fmaf(ax, bx, pc[0]); pc[1] = fmaf(ax, by, pc[1]); pc[2] = fmaf(ax, bz, pc[2]);
    pc[3] = fmaf(ay, bx, pc[3]); pc[4] = fmaf(ay, by, pc[4]); pc[5] = fmaf(ay, bz, pc[5]);
    pc[6] = fmaf(az, bx, pc[6]); pc[7] = fmaf(az, by, pc[7]); pc[8] = fmaf(az, bz, pc[8]);
  }
  float cov[3][3];
  for (int k = 0; k < 9; ++k) cov[k / 3][k % 3] = block_sum256(pc[k], sbuf);

  if (t == 0) {
    // Jacobi eigen of S = cov^T cov -> V, singular values; U = cov V / sigma
    float Sm[3][3];
    for (int r2 = 0; r2 < 3; ++r2)
      for (int c2 = 0; c2 < 3; ++c2) {
        float s = 0.f;
        for (int k = 0; k < 3; ++k) s += cov[k][r2] * cov[k][c2];
        Sm[r2][c2] = s;
      }
    float V[3][3] = {{1.f, 0.f, 0.f}, {0.f, 1.f, 0.f}, {0.f, 0.f, 1.f}};
    for (int sweep = 0; sweep < 16; ++sweep) {
      for (int pi = 0; pi < 3; ++pi) {
        const int p = (pi == 2) ? 1 : 0;
        const int q = (pi == 0) ? 1 : 2;
        float apq = Sm[p][q];
        if (fabsf(apq) < 1e-24f) continue;
        float tau = (Sm[q][q] - Sm[p][p]) / (2.f * apq);
        float t2  = (tau >= 0.f ? 1.f : -1.f) / (fabsf(tau) + sqrtf(1.f + tau * tau));
        float c   = 1.f / sqrtf(1.f + t2 * t2);
        float s   = t2 * c;
        for (int k = 0; k < 3; ++k) {
          float kp = Sm[k][p], kq = Sm[k][q];
          Sm[k][p] = c * kp - s * kq; Sm[k][q] = s * kp + c * kq;
        }
        for (int k = 0; k < 3; ++k) {
          float pk = Sm[p][k], qk = Sm[q][k];
          Sm[p][k] = c * pk - s * qk; Sm[q][k] = s * pk + c * qk;
        }
        for (int k = 0; k < 3; ++k) {
          float vp = V[k][p], vq = V[k][q];
          V[k][p] = c * vp - s * vq; V[k][q] = s * vp + c * vq;
        }
      }
    }
    float dv[3] = {Sm[0][0], Sm[1][1], Sm[2][2]};
    for (int a2 = 0; a2 < 2; ++a2)
      for (int b2 = a2 + 1; b2 < 3; ++b2)
        if (dv[b2] > dv[a2]) {
          float td = dv[a2]; dv[a2] = dv[b2]; dv[b2] = td;
          for (int k = 0; k < 3; ++k) { float tv = V[k][a2]; V[k][a2] = V[k][b2]; V[k][b2] = tv; }
        }
    float U[3][3];
    for (int k = 0; k < 3; ++k) {
      float sg = sqrtf(fmaxf(dv[k], 0.f));
      float ux = cov[0][0] * V[0][k] + cov[0][1] * V[1][k] + cov[0][2] * V[2][k];
      float uy = cov[1][0] * V[0][k] + cov[1][1] * V[1][k] + cov[1][2] * V[2][k];
      float uz = cov[2][0] * V[0][k] + cov[2][1] * V[1][k] + cov[2][2] * V[2][k];
      float nrm = sqrtf(ux * ux + uy * uy + uz * uz);
      if (sg > 1e-20f && nrm > 1e-20f) {
        float in2 = 1.f / nrm;
        U[0][k] = ux * in2; U[1][k] = uy * in2; U[2][k] = uz * in2;
      } else {
        U[0][k] = 0.f; U[1][k] = 0.f; U[2][k] = 0.f;
      }
    }
    float n2 = U[0][2] * U[0][2] + U[1][2] * U[1][2] + U[2][2] * U[2][2];
    if (n2 < 0.5f) {  // degenerate smallest direction: rebuild from cross product
      float cx = U[1][0] * U[2][1] - U[2][0] * U[1][1];
      float cy = U[2][0] * U[0][1] - U[0][0] * U[2][1];
      float cz = U[0][0] * U[1][1] - U[1][0] * U[0][1];
      float nn = sqrtf(cx * cx + cy * cy + cz * cz) + 1e-20f;
      U[0][2] = cx / nn; U[1][2] = cy / nn; U[2][2] = cz / nn;
    }
    float Rp[3][3];
    for (int r2 = 0; r2 < 3; ++r2)
      for (int c2 = 0; c2 < 3; ++c2)
        Rp[r2][c2] = V[r2][0] * U[c2][0] + V[r2][1] * U[c2][1] + V[r2][2] * U[c2][2];
    float det = Rp[0][0] * (Rp[1][1] * Rp[2][2] - Rp[1][2] * Rp[2][1])
              - Rp[0][1] * (Rp[1][0] * Rp[2][2] - Rp[1][2] * Rp[2][0])
              + Rp[0][2] * (Rp[1][0] * Rp[2][1] - Rp[1][1] * Rp[2][0]);
    float Rf[3][3];
    for (int r2 = 0; r2 < 3; ++r2)
      for (int c2 = 0; c2 < 3; ++c2)
        Rf[r2][c2] = (det > 0.f) ? Rp[r2][c2] : (Rp[r2][c2] - 2.f * V[r2][2] * U[c2][2]);
    for (int r2 = 0; r2 < 3; ++r2)
      for (int c2 = 0; c2 < 3; ++c2)
        outR[b * 9 + r2 * 3 + c2] = Rf[r2][c2];
    for (int r2 = 0; r2 < 3; ++r2)
      outT[b * 3 + r2] =
          -(Rf[r2][0] * ca[0] + Rf[r2][1] * ca[1] + Rf[r2][2] * ca[2]) + cb[r2];
  }
}

// ---------------------------------------------------------------- launch
extern "C" void kernel_launch(void* const* d_in, const int* in_sizes, int n_in,
                              void* d_out, int out_size, void* d_ws, size_t ws_size,
                              hipStream_t stream) {
  const int B = 16, D = 256, M = 2048, N = 2048;
  const int CCH = 8;  // column-matvec row chunks

  const float* se  = (const float*)d_in[0];
  const float* te  = (const float*)d_in[1];
  const float* src = (const float*)d_in[2];
  const float* tgt = (const float*)d_in[3];
  const float* rho = (const float*)d_in[4];

  float* out   = (float*)d_out;
  float* R_out = out;
  float* t_out = out + (size_t)B * 9;
  float* Kmat  = out + (size_t)B * 9 + (size_t)B * 3;  // 268 MB matches region, used as K0 scratch

  float* ws = (float*)d_ws;
  size_t off = 0;
  float* u      = ws + off; off += (size_t)B * M;
  float* v      = ws + off; off += (size_t)B * N;
  float* y      = ws + off; off += (size_t)B * M;
  float* z      = ws + off; off += (size_t)B * N;
  float* lse    = ws + off; off += (size_t)B * M;
  float* rowp   = ws + off; off += (size_t)B * M;
  float* rowsum = ws + off; off += (size_t)B * M;
  float* wref   = ws + off; off += (size_t)B * M * 3;
  float* zpart  = ws + off; off += (size_t)B * CCH * N;
  float* Spart  = ws + off; off += (size_t)B * (N / 256);
  float* g      = ws + off; off += (size_t)B;

  // 1) scores -> Kmat (WMMA f32 GEMM with LDS staging, scaled by 1/sqrt(D))
  gemm_scores_k<<<dim3(M / 64, N / 64, B), 128, 0, stream>>>(se, te, Kmat, D, M, N, 0.0625f);
  // 2) row logsumexp
  row_lse_k<<<(B * M) / 8, 256, 0, stream>>>(Kmat, lse, N);
  // 3) K0 = exp((s - lse)/rho), per-row partial sums; init u=v=1; g = 1/clip(sum K0)
  const int initn = (B * M > B * N) ? B * M : B * N;
  init_uv_k<<<(initn + 255) / 256, 256, 0, stream>>>(u, v, B * M, B * N);
  exp_rows_k<<<B * M, 256, 0, stream>>>(Kmat, lse, rho, rowp, N);
  g_init_k<<<B, 256, 0, stream>>>(rowp, g, M);
  // 4) 5 Sinkhorn iterations as diagonal-scaling updates (K = g * u o K0 o v)
  for (int it = 0; it < 5; ++it) {
    rowmv_k<<<(B * M) / 8, 256, 0, stream>>>(Kmat, v, y, M, N);
    u_update_k<<<(B * M + 255) / 256, 256, 0, stream>>>(u, y, g, 1.f / (float)M, M, B * M);
    colmv_part_k<<<dim3(N / 256, CCH, B), 256, 0, stream>>>(Kmat, u, zpart, M, N, CCH);
    z_reduce_k<<<(B * N + 255) / 256, 256, 0, stream>>>(zpart, z, N, CCH, B * N);
    v_update_k<<<dim3(N / 256, B), 256, 0, stream>>>(v, z, g, Spart, 1.f / (float)N, N);
    g_update_k<<<1, B, 0, stream>>>(g, Spart, N / 256);
  }
  // 5) materialize matches in place; row sums + weighted_ref = matches @ t^T / rowsum
  finalize_rows_k<<<(B * M) / 8, 256, 0, stream>>>(Kmat, u, v, g, tgt, rowsum, wref, M, N);
  // 6) weighted rigid fit (3x3 SVD) per batch
  rigid_k<<<B, 256, 0, stream>>>(src, wref, rowsum, R_out, t_out, M);
}